// GraphConvReg_6536940224564
// MI455X (gfx1250) — compile-verified
//
#include <hip/hip_runtime.h>

// GraphConv + reg loss for MI455X (gfx1250, wave32).
// Pipeline on one stream:
//   1) zero y (d_out) and reduction scratch
//   2) WMMA f32 GEMM: h = x @ W^T + b          (v_wmma_f32_16x16x4_f32)
//   3) edge scatter: y[dst] += h[src]*w, accumulating colsum_c = sum_e w_e h[src_e,c]
//   4) column reduce over h: T_c = sum_n mu_n h_nc, S1 = sum (mu_n h_nc)^2
//   5) finalize: reg = (S1 - 2*sum_c mx_c T_c + N*sum_c mx_c^2) / (N*C), mx_c = colsum_c/N
// d_ws layout: h[N*32] floats, then colsum[32], T[32], S1[1].

#define FDIM 256
#define CDIM 32

typedef float v2f __attribute__((ext_vector_type(2)));
typedef float v8f __attribute__((ext_vector_type(8)));

__global__ void __launch_bounds__(256)
zero_kernel(float* __restrict__ y, float* __restrict__ accs, long long ny) {
    long long i = (long long)blockIdx.x * blockDim.x + threadIdx.x;
    const long long stride = (long long)gridDim.x * blockDim.x;
    if (i < 80) accs[i] = 0.0f;   // colsum[32] + T[32] + S1[1] (padded)
    for (; i < ny; i += stride) y[i] = 0.0f;
}

// h = x @ W^T + bias, fp32 WMMA. One wave = 16 rows x 32 cols (two 16x16 tiles).
// A-fragment (16x4 f32): lane&15 = M, (lane>>4)*2 = K pair -> float2 per lane.
// B-fragment (4x16 f32): lane&15 = N (i.e. class), (lane>>4)*2 = K pair; B[k,n] = W[n,k].
__global__ void __launch_bounds__(256)
gemm_kernel(const float* __restrict__ x, const float* __restrict__ Wt,
            const float* __restrict__ bias, float* __restrict__ h, int nTiles) {
    const int lane = threadIdx.x & 31;
    const int wid  = threadIdx.x >> 5;
    const int tile = blockIdx.x * 8 + wid;
    if (tile >= nTiles) return;                 // no syncthreads: uniform wave exit ok
    const int rb   = tile * 16;
    const int m    = lane & 15;
    const int koff = (lane >> 4) * 2;

    const float* xa  = x  + (size_t)(rb + m) * FDIM + koff;   // row of x, even K pairs
    const float* wb0 = Wt + (size_t)m        * FDIM + koff;   // W rows 0..15  (cols 0..15)
    const float* wb1 = Wt + (size_t)(m + 16) * FDIM + koff;   // W rows 16..31 (cols 16..31)

    v8f acc0 = {}; v8f acc1 = {};
#pragma unroll 8
    for (int kk = 0; kk < FDIM / 4; ++kk) {
        v2f a  = *(const v2f*)(xa  + 4 * kk);
        v2f b0 = *(const v2f*)(wb0 + 4 * kk);
        v2f b1 = *(const v2f*)(wb1 + 4 * kk);
        acc0 = __builtin_amdgcn_wmma_f32_16x16x4_f32(false, a, false, b0,
                                                     (short)0, acc0, false, false);
        acc1 = __builtin_amdgcn_wmma_f32_16x16x4_f32(false, a, false, b1,
                                                     (short)0, acc1, false, false);
    }

    // D layout: VGPR v holds row (v) for lanes 0-15 and row (v+8) for lanes 16-31; N = lane&15.
    const int   rowadd = (lane >> 4) * 8;
    const float b0s = bias[m];
    const float b1s = bias[m + 16];
#pragma unroll
    for (int v = 0; v < 8; ++v) {
        const size_t row = (size_t)(rb + v + rowadd);
        h[row * CDIM + m]        = acc0[v] + b0s;
        h[row * CDIM + m + 16]   = acc1[v] + b1s;
    }
}

// One wave per edge iteration: lane = class. 128B coalesced gather of h[src],
// 128B contiguous f32 atomics into y[dst] (L2-resident). Also accumulates
// colsum_c = sum_e w_e * h[src_e, c] with one atomic per lane per wave.
__global__ void __launch_bounds__(256)
scatter_kernel(const float* __restrict__ h, const float* __restrict__ w,
               const int* __restrict__ src, const int* __restrict__ dst,
               float* __restrict__ y, float* __restrict__ colsum, int E_) {
    const int lane = threadIdx.x & 31;
    const int wid  = (blockIdx.x * blockDim.x + threadIdx.x) >> 5;
    const int nw   = (gridDim.x * blockDim.x) >> 5;
    float acc = 0.0f;
    for (int e = wid; e < E_; e += nw) {
        const int   s  = src[e];
        const int   d  = dst[e];
        const float we = w[e];
        const float msg = h[(size_t)s * CDIM + lane] * we;
        atomicAdd(&y[(size_t)d * CDIM + lane], msg);
        acc += msg;
    }
    atomicAdd(&colsum[lane], acc);
}

// T_c = sum_n mu_n h_nc ; S1 = sum_{n,c} (mu_n h_nc)^2 ; mu_n = u_sum[n]/N
__global__ void __launch_bounds__(256)
reduce_kernel(const float* __restrict__ h, const float* __restrict__ u_sum,
              float* __restrict__ T, float* __restrict__ S1, int N_, float invN) {
    const int lane = threadIdx.x & 31;
    const int wid  = (blockIdx.x * blockDim.x + threadIdx.x) >> 5;
    const int nw   = (gridDim.x * blockDim.x) >> 5;
    float accT = 0.0f, accS = 0.0f;
    for (int n = wid; n < N_; n += nw) {
        const float mu = u_sum[n] * invN;
        const float t  = mu * h[(size_t)n * CDIM + lane];
        accT += t;
        accS += t * t;
    }
    atomicAdd(&T[lane], accT);
    for (int off = 16; off > 0; off >>= 1) accS += __shfl_down(accS, off, 32);
    if (lane == 0) atomicAdd(S1, accS);
}

__global__ void
finalize_kernel(const float* __restrict__ colsum, const float* __restrict__ T,
                const float* __restrict__ S1, float* __restrict__ out_reg, float fN) {
    const int lane = threadIdx.x;                // 32 threads
    const float mx = colsum[lane] / fN;
    float term = fN * mx * mx - 2.0f * mx * T[lane];
    for (int off = 16; off > 0; off >>= 1) term += __shfl_down(term, off, 32);
    if (lane == 0) *out_reg = (term + *S1) / (fN * (float)CDIM);
}

extern "C" void kernel_launch(void* const* d_in, const int* in_sizes, int n_in,
                              void* d_out, int out_size, void* d_ws, size_t ws_size,
                              hipStream_t stream) {
    const float* x     = (const float*)d_in[0];   // [N, 256]
    const float* w     = (const float*)d_in[1];   // [E]
    const float* u_sum = (const float*)d_in[2];   // [N]
    const float* Wt    = (const float*)d_in[3];   // [32, 256]
    const float* bias  = (const float*)d_in[4];   // [32]
    const int*   src   = (const int*)d_in[5];     // [E]
    const int*   dst   = (const int*)d_in[6];     // [E]
    const int N_ = in_sizes[2];
    const int E_ = in_sizes[1];

    float* y      = (float*)d_out;                // [N, 32] then reg scalar
    float* h      = (float*)d_ws;                 // [N, 32]
    float* accs   = h + (size_t)N_ * CDIM;        // colsum[32], T[32], S1[1]
    float* colsum = accs;
    float* T      = accs + 32;
    float* S1     = accs + 64;

    zero_kernel<<<4096, 256, 0, stream>>>(y, accs, (long long)N_ * CDIM);

    const int nTiles = (N_ + 15) / 16;            // N % 16 == 0 for this problem
    gemm_kernel<<<(nTiles + 7) / 8, 256, 0, stream>>>(x, Wt, bias, h, nTiles);

    scatter_kernel<<<2048, 256, 0, stream>>>(h, w, src, dst, y, colsum, E_);

    reduce_kernel<<<1024, 256, 0, stream>>>(h, u_sum, T, S1, N_, 1.0f / (float)N_);

    finalize_kernel<<<1, 32, 0, stream>>>(colsum, T, S1, y + (size_t)N_ * CDIM, (float)N_);
}